// Layer_75153337745803
// MI455X (gfx1250) — compile-verified
//
#include <hip/hip_runtime.h>
#include <hip/hip_bf16.h>
#include <math.h>

// ---------------- problem constants (from reference) ----------------
constexpr int kB = 64;        // batch rows (M)
constexpr int kD = 2048;      // hidden (K)
constexpr int kV = 128000;    // vocab (N)
constexpr float kTopP = 0.9f;
constexpr int kHardMask = 162;
constexpr float kNeg = -1e9f;

// ---------------- WMMA / TDM vector types (wave32, gfx1250) ----------------
typedef float v2f  __attribute__((ext_vector_type(2)));
typedef float v8f  __attribute__((ext_vector_type(8)));
typedef unsigned int u32x4 __attribute__((ext_vector_type(4)));
typedef int          i32x8 __attribute__((ext_vector_type(8)));
typedef int          i32x4 __attribute__((ext_vector_type(4)));

// ---------------- GEMM tiling ----------------
constexpr int NBLK = 128;     // N columns per block
constexpr int KC   = 128;     // K chunk staged in LDS
constexpr int LDA  = KC + 4;  // padded LDS row stride (floats); 132%64=4 -> conflict-free

// LDS offset (addrspace(3) pointers are 32-bit on AMDGPU)
__device__ __forceinline__ unsigned lds_offset_of(const void* p) {
    return (unsigned)(uintptr_t)(__attribute__((address_space(3))) const void*)p;
}

// =====================================================================
// Kernel 1: logits = A @ W + b, hard-mask col 162.
// grid = (kV/NBLK) blocks, 256 threads (8 waves). Wave w:
//   m-tile = w>>1 (16 rows), n-half = w&1 (64 cols = 4 subtiles of 16).
// A chunks staged to LDS via the Tensor Data Mover (tensor_load_to_lds,
// D# pads +4 DWORDs per 128 -> stride 132); W streamed with 32-bit
// saddr+scale_offset global loads; math on V_WMMA_F32_16X16X4_F32.
// =====================================================================
__global__ __launch_bounds__(256)
void gemm_logits_wmma(const float* __restrict__ A,
                      const float* __restrict__ W,
                      const float* __restrict__ bias,
                      float* __restrict__ logits) {
    __shared__ float As[kB * LDA];

    const int tid  = threadIdx.x;
    const int wid  = tid >> 5;
    const int lane = tid & 31;
    const int lr   = lane & 15;          // row/col-within-16 index
    const int koff = (lane >> 4) << 1;   // 0 for lanes 0-15, 2 for lanes 16-31

    const int m0     = (wid >> 1) << 4;                    // 0,16,32,48
    const int n_base = blockIdx.x * NBLK + ((wid & 1) << 6);

    v8f acc[4];
#pragma unroll
    for (int j = 0; j < 4; ++j) acc[j] = (v8f)0.0f;

    // 32-bit W element index base (per-lane); stays < 2^31 for all k.
    const int wbase = koff * kV + n_base + lr;

    // ---- TDM descriptor pieces that do not change across chunks ----
    const unsigned ldsa = lds_offset_of(As);
    // group1: data_size=4B(2), pad_enable, pad_interval=6 (128 DW),
    // pad_amount=3 (4 DW); tensor_dim0=kD, tensor_dim1=kB; tile 128x64;
    // tensor_dim0_stride=kD.
    const unsigned g1w0 = (2u << 16) | (1u << 20) | (6u << 22) | (3u << 25);
    const unsigned g1w1 = ((unsigned)kD & 0xFFFFu) << 16;          // dim0[15:0]
    const unsigned g1w2 = ((unsigned)kB & 0xFFFFu) << 16;          // dim0[31:16]=0, dim1[15:0]
    const unsigned g1w3 = ((unsigned)KC & 0xFFFFu) << 16;          // dim1[31:16]=0, tile_dim0
    const unsigned g1w4 = (unsigned)kB;                            // tile_dim1, tile_dim2=0
    const unsigned g1w5 = (unsigned)kD;                            // dim0_stride[31:0]
    const i32x8 g1 = { (int)g1w0, (int)g1w1, (int)g1w2, (int)g1w3,
                       (int)g1w4, (int)g1w5, 0, 0 };
    const i32x4 gz4 = { 0, 0, 0, 0 };      // groups 2/3 unused (2D tensor)
    const i32x8 gz8 = { 0, 0, 0, 0, 0, 0, 0, 0 };

    for (int k0 = 0; k0 < kD; k0 += KC) {
        // ---- TDM: stage A[0:64, k0:k0+KC] -> LDS (wave 0 issues) ----
        if (wid == 0) {
            const unsigned long long ga =
                (unsigned long long)(uintptr_t)A + (unsigned long long)k0 * 4ull;
            const unsigned ga_lo = (unsigned)ga;
            const unsigned ga_hi = (unsigned)(ga >> 32) & 0x1FFFFFFu;
            const u32x4 g0 = { 1u /*count=1*/, ldsa, ga_lo, ga_hi | (2u << 30) /*type=image*/ };
            __builtin_amdgcn_tensor_load_to_lds(g0, g1, gz4, gz4, gz8, 0);
            __builtin_amdgcn_s_wait_tensorcnt(0);
        }
        __syncthreads();

#pragma unroll 4
        for (int kk = 0; kk < KC; kk += 4) {
            const int wi = wbase + (k0 + kk) * kV;   // 32-bit index -> saddr form
            // prefetch W ~16 K-rows ahead (global_prefetch_b8)
            __builtin_prefetch((const void*)&W[wi + 16 * kV], 0, 1);

            // A fragment 16x4: lanes 0-15 -> K=kk+0,1 ; lanes 16-31 -> K=kk+2,3
            const v2f a = *(const v2f*)&As[(m0 + lr) * LDA + kk + koff];

#pragma unroll
            for (int j = 0; j < 4; ++j) {
                v2f b;
                b.x = W[wi + j * 16];        // W[k+koff  ][n]
                b.y = W[wi + j * 16 + kV];   // W[k+koff+1][n]
                acc[j] = __builtin_amdgcn_wmma_f32_16x16x4_f32(
                    /*neg_a=*/false, a, /*neg_b=*/false, b,
                    /*c_mod=*/(short)0, acc[j],
                    /*reuse_a=*/false, /*reuse_b=*/false);
            }
        }
        __syncthreads();
    }

    // ---- epilogue: bias + hard mask, store C tiles (32-bit indices) ----
    const int rhi = (lane >> 4) << 3;   // +8 rows for lanes 16-31
#pragma unroll
    for (int j = 0; j < 4; ++j) {
        const int col = n_base + j * 16 + lr;
        const float bv = bias[col];
#pragma unroll
        for (int vv = 0; vv < 8; ++vv) {
            const int row = m0 + vv + rhi;
            float val = acc[j][vv] + bv;
            if (col == kHardMask) val = kNeg;
            logits[row * kV + col] = val;
        }
    }
}

// =====================================================================
// Kernel 2: per-row max and sum(exp(x-max))
// =====================================================================
__global__ __launch_bounds__(256)
void rowstats_kernel(const float* __restrict__ logits,
                     float* __restrict__ rowmax, float* __restrict__ rowsum) {
    const int row = blockIdx.x, tid = threadIdx.x;
    __shared__ float red[256];
    const float* rp = logits + (size_t)row * kV;

    float m = -3.4e38f;
    for (int i = tid; i < kV; i += 256) m = fmaxf(m, rp[i]);
    red[tid] = m; __syncthreads();
    for (int s = 128; s > 0; s >>= 1) { if (tid < s) red[tid] = fmaxf(red[tid], red[tid + s]); __syncthreads(); }
    const float rm = red[0]; __syncthreads();

    float sum = 0.f;
    for (int i = tid; i < kV; i += 256) sum += __expf(rp[i] - rm);
    red[tid] = sum; __syncthreads();
    for (int s = 128; s > 0; s >>= 1) { if (tid < s) red[tid] += red[tid + s]; __syncthreads(); }
    if (tid == 0) { rowmax[row] = rm; rowsum[row] = red[0]; }
}

// =====================================================================
// Kernel 3: probs in place (logits buffer becomes probs); grid=(kV/256, kB)
// =====================================================================
__global__ __launch_bounds__(256)
void probs_kernel(float* __restrict__ logits,
                  const float* __restrict__ rowmax, const float* __restrict__ rowsum) {
    const int row = blockIdx.y;
    const int i = blockIdx.x * 256 + threadIdx.x;
    float* rp = logits + (size_t)row * kV;
    rp[i] = __expf(rp[i] - rowmax[row]) / rowsum[row];
}

// =====================================================================
// Kernel 4/5/6: descending bucket "sort" via 4096-bin histogram of the
// monotone float-bit key (probs >= 0), exclusive scan, then scatter.
// =====================================================================
__global__ void zero_hist_kernel(unsigned* __restrict__ hist) {
    hist[blockIdx.x * 256 + threadIdx.x] = 0u;
}

__device__ __forceinline__ int desc_bin(float p) {
    unsigned key = __float_as_uint(p);          // monotone for p >= 0
    int b = 4095 - (int)(key >> 19);
    return b < 0 ? 0 : (b > 4095 ? 4095 : b);
}

__global__ __launch_bounds__(256)
void hist_kernel(const float* __restrict__ probs, unsigned* __restrict__ hist) {
    const int row = blockIdx.y;
    const int i = blockIdx.x * 256 + threadIdx.x;
    const float p = probs[(size_t)row * kV + i];
    atomicAdd(&hist[row * 4096 + desc_bin(p)], 1u);
}

__global__ void scan_bins_kernel(unsigned* __restrict__ hist) {
    const int row = threadIdx.x;
    if (row < kB) {
        unsigned* h = hist + row * 4096;
        unsigned run = 0;
        for (int b = 0; b < 4096; ++b) { unsigned c = h[b]; h[b] = run; run += c; }
    }
}

__global__ __launch_bounds__(256)
void scatter_kernel(const float* __restrict__ probs, unsigned* __restrict__ cursor,
                    float* __restrict__ sortedp, int* __restrict__ order) {
    const int row = blockIdx.y;
    const int i = blockIdx.x * 256 + threadIdx.x;
    const float p = probs[(size_t)row * kV + i];
    const unsigned pos = atomicAdd(&cursor[row * 4096 + desc_bin(p)], 1u);
    sortedp[(size_t)row * kV + pos] = p;
    order[(size_t)row * kV + pos] = i;
}

// =====================================================================
// Kernel 7: per-row chunked inclusive scan over sorted probs, apply
// keep = cum <= TOP_P, zero the rest, then renormalize in place.
// =====================================================================
__global__ __launch_bounds__(1024)
void topp_scan_kernel(float* __restrict__ sortedp, float* __restrict__ keptsum) {
    const int row = blockIdx.x, tid = threadIdx.x;
    __shared__ float s[1024];
    __shared__ float sh_carry, sh_kept;
    if (tid == 0) { sh_carry = 0.f; sh_kept = 0.f; }
    __syncthreads();

    float* rp = sortedp + (size_t)row * kV;
    for (int base = 0; base < kV; base += 1024) {
        const float v = rp[base + tid];
        s[tid] = v; __syncthreads();
        for (int off = 1; off < 1024; off <<= 1) {
            const float t = (tid >= off) ? s[tid - off] : 0.f;
            __syncthreads();
            s[tid] += t; __syncthreads();
        }
        const float cum = sh_carry + s[tid];
        const float tot = s[1023];
        const bool keep = (cum <= kTopP);
        rp[base + tid] = keep ? v : 0.f;
        __syncthreads();
        s[tid] = keep ? cum : 0.f; __syncthreads();        // max kept cum (prefix property)
        for (int q = 512; q > 0; q >>= 1) { if (tid < q) s[tid] = fmaxf(s[tid], s[tid + q]); __syncthreads(); }
        if (tid == 0) { sh_kept = fmaxf(sh_kept, s[0]); sh_carry += tot; }
        __syncthreads();
    }

    const float ks = sh_kept;
    const float inv = 1.f / (ks > 0.f ? ks : 1.f);
    for (int base = 0; base < kV; base += 1024) rp[base + tid] *= inv;
    if (tid == 0) keptsum[row] = ks;
}

// =====================================================================
// Kernel 8: inverse-CDF sample over the renormalized top-p distribution.
// =====================================================================
__global__ __launch_bounds__(1024)
void sample_kernel(const float* __restrict__ p, const int* __restrict__ order,
                   float* __restrict__ tokens) {
    const int row = blockIdx.x, tid = threadIdx.x;
    __shared__ float s[1024];
    __shared__ float sh_carry; __shared__ int sh_idx;
    if (tid == 0) { sh_carry = 0.f; sh_idx = 0x7fffffff; }
    __syncthreads();

    unsigned h = 42u ^ (unsigned)(row * 0x9E3779B9u);
    h ^= h >> 16; h *= 0x7feb352du; h ^= h >> 15; h *= 0x846ca68bu; h ^= h >> 16;
    const float u = (float)(h >> 8) * (1.0f / 16777216.0f);

    const float* rp = p + (size_t)row * kV;
    for (int base = 0; base < kV; base += 1024) {
        const float v = rp[base + tid];
        s[tid] = v; __syncthreads();
        for (int off = 1; off < 1024; off <<= 1) {
            const float t = (tid >= off) ? s[tid - off] : 0.f;
            __syncthreads();
            s[tid] += t; __syncthreads();
        }
        const float cum = sh_carry + s[tid];
        if (cum >= u && (cum - v) < u) atomicMin(&sh_idx, base + tid);
        const float tot = s[1023];
        __syncthreads();
        if (tid == 0) sh_carry += tot;
        __syncthreads();
    }
    if (tid == 0) {
        const int idx = (sh_idx == 0x7fffffff) ? (kV - 1) : sh_idx;
        tokens[row] = (float)order[(size_t)row * kV + idx];
    }
}

// =====================================================================
// launcher
// =====================================================================
extern "C" void kernel_launch(void* const* d_in, const int* in_sizes, int n_in,
                              void* d_out, int out_size, void* d_ws, size_t ws_size,
                              hipStream_t stream) {
    const float* A    = (const float*)d_in[0];   // [64, 2048]
    const float* W    = (const float*)d_in[1];   // [2048, 128000]
    const float* bias = (const float*)d_in[2];   // [128000]

    // workspace layout
    float*    logits = (float*)d_ws;                        // kB*kV floats (-> probs in place)
    int*      order  = (int*)(logits + (size_t)kB * kV);    // kB*kV ints
    unsigned* hist   = (unsigned*)(order + (size_t)kB * kV);// kB*4096
    float*    rowmax = (float*)(hist + (size_t)kB * 4096);  // kB
    float*    rowsum = rowmax + kB;                         // kB
    float*    keptsum= rowsum + kB;                         // kB

    float* tok_out = (float*)d_out;       // [64] tokens (as float)
    float* p_out   = tok_out + kB;        // [64, 128000] sorted/renormalized p

    // 1) GEMM + bias + hard mask -> logits
    gemm_logits_wmma<<<kV / NBLK, 256, 0, stream>>>(A, W, bias, logits);
    // 2) softmax stats
    rowstats_kernel<<<kB, 256, 0, stream>>>(logits, rowmax, rowsum);
    // 3) probs in place
    probs_kernel<<<dim3(kV / 256, kB), 256, 0, stream>>>(logits, rowmax, rowsum);
    // 4-6) descending bucket sort: hist -> scan -> scatter (into d_out p region)
    zero_hist_kernel<<<(kB * 4096) / 256, 256, 0, stream>>>(hist);
    hist_kernel<<<dim3(kV / 256, kB), 256, 0, stream>>>(logits, hist);
    scan_bins_kernel<<<1, kB, 0, stream>>>(hist);
    scatter_kernel<<<dim3(kV / 256, kB), 256, 0, stream>>>(logits, hist, p_out, order);
    // 7) cumsum + top-p filter + renormalize (in place in d_out)
    topp_scan_kernel<<<kB, 1024, 0, stream>>>(p_out, keptsum);
    // 8) sample token per row
    sample_kernel<<<kB, 1024, 0, stream>>>(p_out, order, tok_out);
}